// Conv2d_8907762171992
// MI455X (gfx1250) — compile-verified
//
#include <hip/hip_runtime.h>
#include <hip/hip_bf16.h>

// 7x7 VALID fp32 conv on 4096x4096 via V_WMMA_F32_16X16X4_F32 (CDNA5).
// Each wave32 computes a 16x16 output tile:
//   D[r][c] = bias + sum_kh sum_j X[row0+r+kh][col0+j] * T_kh[j][c]
// where T_kh is the 22x16 banded Toeplitz matrix of filter row kh
// (T[j][c] = w[kh][j-c] for 0<=j-c<=6, else 0), K padded 22->24,
// executed as 7*6 = 42 chained 16x16x4 fp32 WMMAs.
//
// B fragments are tile-invariant: precomputed once per workgroup into an LDS
// table (42 chunks x 32 lanes x float2), fetched with one ds_load_b64 at an
// immediate offset. A fragments are global_load_b64 at immediate offsets off
// a single per-lane base pointer. Edge tiles are shifted inward (overlapping
// stores write identical values), so there are no per-chunk clamps and no
// store predicates.

#define H   4096
#define W   4096
#define KHH 7
#define KWW 7
#define OH  (H - KHH + 1)   // 4090
#define OW  (W - KWW + 1)   // 4090

#define NCHUNK (KHH * 6)    // 42 K=4 chunks

typedef __attribute__((ext_vector_type(2))) float v2f;
typedef __attribute__((ext_vector_type(8))) float v8f;

__global__ __launch_bounds__(256) void conv7x7_wmma_f32(
    const float* __restrict__ x,
    const float* __restrict__ w,
    const float* __restrict__ bias,
    float* __restrict__ out)
{
    // B-fragment table: [chunk][lane] -> float2  (10.5 KB)
    __shared__ float blds[NCHUNK * 32 * 2];

    const int tid = threadIdx.x;

    // ---- One-time per workgroup: build the banded-Toeplitz B fragments ----
    // chunk = kh*6+q covers K rows jbase..jbase+3; lane n = output column.
    // Lanes 0-15 hold K rows {jbase, jbase+1}, lanes 16-31 {jbase+2, jbase+3}.
    for (int e = tid; e < NCHUNK * 32; e += 256) {
        const int chunk = e >> 5;
        const int l     = e & 31;
        const int kh    = chunk / 6;
        const int q     = chunk - 6 * kh;
        const int lm    = l & 15;
        const int jb    = q * 4 + (l >> 4) * 2;
        const int d0    = jb - lm;
        const int d1    = d0 + 1;
        const bool m0   = ((unsigned)d0 <= 6u);
        const bool m1   = ((unsigned)d1 <= 6u);
        const float w0  = w[m0 ? (kh * KWW + d0) : 0];   // index kept in-range
        const float w1  = w[m1 ? (kh * KWW + d1) : 0];
        blds[2 * e]     = m0 ? w0 : 0.0f;
        blds[2 * e + 1] = m1 ? w1 : 0.0f;
    }
    __syncthreads();

    const int lane  = tid & 31;
    const int laneM = lane & 15;
    const int half  = lane >> 4;
    const int wave  = tid >> 5;

    // Tile origin, shifted inward at the edges so every load/store is
    // in-bounds (overlap regions get identical values -> deterministic).
    int row0 = blockIdx.y * 16;
    if (row0 > OH - 16) row0 = OH - 16;
    int col0 = blockIdx.x * 128 + wave * 16;
    if (col0 > OW - 16) col0 = OW - 16;

    // Per-lane A base: row (row0+laneM), columns start at col0 + 2*half.
    // Chunk (kh,q) then lives at constant byte offset kh*W*4 + q*16.
    const size_t rowbase = (size_t)(row0 + laneM) * (size_t)W;
    const float* __restrict__ abase = x + rowbase + (col0 + 2 * half);
    // Chunk q=5, upper-half lanes read j=22,23 (zero B rows) -> clamp base.
    int c5 = col0 + 20 + 2 * half;
    if (c5 > W - 2) c5 = W - 2;
    const float* __restrict__ abase5 = x + rowbase + c5;

    const float* __restrict__ bbase = blds + 2 * lane;

    const float bv = bias[0];
    v8f acc0, acc1;   // even/odd kh accumulators: break the WMMA RAW chain
#pragma unroll
    for (int i = 0; i < 8; ++i) { acc0[i] = bv; acc1[i] = 0.0f; }

#pragma unroll
    for (int kh = 0; kh < KHH; ++kh) {
        v8f acc = (kh & 1) ? acc1 : acc0;
#pragma unroll
        for (int q = 0; q < 6; ++q) {
            // A fragment: immediate-offset global_load_b64
            v2f a = (q < 5) ? *(const v2f*)(abase + (size_t)kh * W + 4 * q)
                            : *(const v2f*)(abase5 + (size_t)kh * W);
            // B fragment: immediate-offset ds_load_b64
            v2f b = *(const v2f*)(bbase + (kh * 6 + q) * 64);

            acc = __builtin_amdgcn_wmma_f32_16x16x4_f32(
                /*neg_a=*/false, a, /*neg_b=*/false, b,
                /*c_mod=*/(short)0, acc,
                /*reuse_a=*/false, /*reuse_b=*/false);
        }
        if (kh & 1) acc1 = acc; else acc0 = acc;
    }

    // ---- Store D: VGPR v -> row (v + 8*half), lane -> column. ----
    // Unconditional, immediate row offsets off one per-lane base.
    float* __restrict__ obase =
        out + (size_t)(row0 + 8 * half) * (size_t)OW + (col0 + laneM);
#pragma unroll
    for (int v = 0; v < 8; ++v)
        obase[(size_t)v * OW] = acc0[v] + acc1[v];
}

extern "C" void kernel_launch(void* const* d_in, const int* in_sizes, int n_in,
                              void* d_out, int out_size, void* d_ws, size_t ws_size,
                              hipStream_t stream) {
    (void)in_sizes; (void)n_in; (void)out_size; (void)d_ws; (void)ws_size;
    const float* x    = (const float*)d_in[0];
    const float* w    = (const float*)d_in[1];
    const float* bias = (const float*)d_in[2];
    float* out        = (float*)d_out;

    dim3 grid((OW + 127) / 128, (OH + 15) / 16);   // 32 x 256 workgroups
    conv7x7_wmma_f32<<<grid, dim3(256), 0, stream>>>(x, w, bias, out);
}